// TensorProductTorch_64407329571144
// MI455X (gfx1250) — compile-verified
//
#include <hip/hip_runtime.h>
#include <hip/hip_bf16.h>

// ---------------------------------------------------------------------------
// e3nn tensor product (64x0e+64x1o+64x2e) x (1x0e+1x1o+1x2e) -> l<=3 filter.
// Per sample: Out[64x72] = X1[64x9] @ W[9x72], W = sum_j x2[j]*CG_j.
// Store-bandwidth bound (~340MB / 23.3TB/s ~= 15us). One wave per sample,
// V_WMMA_F32_16X16X4_F32 for the GEMM (M=64,N=80,K=12 padded), LDS strip in
// output-linear order so the flush is ds_load_b128 + global_store_b128.
// CG coefficients computed EXACTLY at compile time (constexpr port of the
// reference's su2_cg + real<->complex basis change).
// ---------------------------------------------------------------------------

typedef __attribute__((ext_vector_type(2))) float v2f;
typedef __attribute__((ext_vector_type(8))) float v8f;

// ------------------------- compile-time CG machinery -----------------------

constexpr double FACT[11] = {1., 1., 2., 6., 24., 120., 720., 5040., 40320., 362880., 3628800.};

constexpr double csqrt(double x) {
    if (x <= 0.0) return 0.0;
    double g = x > 1.0 ? x : 1.0;
    for (int i = 0; i < 30; ++i) {
        double ng = 0.5 * (g + x / g);
        double d = ng - g;
        if (d < 1e-14 && d > -1e-14) { g = ng; break; }
        g = ng;
    }
    return g;
}

constexpr int imax(int a, int b) { return a > b ? a : b; }
constexpr int imin(int a, int b) { return a < b ? a : b; }

// SU(2) Clebsch-Gordan <j1 m1, j2 m2 | j3 m3> (integer l only here)
constexpr double su2_cg(int j1, int m1, int j2, int m2, int j3, int m3) {
    if (m3 != m1 + m2) return 0.0;
    if (m1 < -j1 || m1 > j1 || m2 < -j2 || m2 > j2 || m3 < -j3 || m3 > j3) return 0.0;
    int vmin = imax(imax(-j1 + j2 + m3, -j1 + m1), 0);
    int vmax = imin(imin(j2 + j3 + m1, j3 - j1 + j2), j3 + m3);
    if (vmax < vmin) return 0.0;
    double num = FACT[j3 + j1 - j2] * FACT[j3 - j1 + j2] * FACT[j1 + j2 - j3] *
                 FACT[j3 + m3] * FACT[j3 - m3];
    double den = FACT[j1 + j2 + j3 + 1] * FACT[j1 - m1] * FACT[j1 + m1] *
                 FACT[j2 - m2] * FACT[j2 + m2];
    double C = csqrt((2.0 * j3 + 1.0) * num / den);
    double S = 0.0;
    for (int v = vmin; v <= vmax; ++v) {
        double sgn = ((v + j2 + m2) & 1) ? -1.0 : 1.0;
        S += sgn * (FACT[j2 + j3 + m1 - v] * FACT[j1 - m1 + v]) /
             (FACT[v] * FACT[j3 - j1 + j2 - v] * FACT[j3 + m3 - v] * FACT[v + j1 - j2 - m3]);
    }
    return C * S;
}

struct CD { double re = 0.0, im = 0.0; };
constexpr CD cmul(CD a, CD b) { return {a.re * b.re - a.im * b.im, a.re * b.im + a.im * b.re}; }
constexpr CD conjc(CD a) { return {a.re, -a.im}; }
constexpr CD cscale(CD a, double s) { return {a.re * s, a.im * s}; }

// Column a (real basis index) of the real->complex change-of-basis matrix Q_l,
// including the (-i)^l phase. Each column has <=2 nonzero (complex-row) entries.
constexpr int qcol(int l, int a, int* rows, CD* vals) {
    const double s2i = 1.0 / csqrt(2.0);
    CD ph{1, 0};
    switch (l & 3) { case 0: ph = {1, 0}; break; case 1: ph = {0, -1}; break;
                     case 2: ph = {-1, 0}; break; default: ph = {0, 1}; break; }
    int mr = a - l;
    if (mr == 0) { rows[0] = l; vals[0] = ph; return 1; }
    if (mr > 0) {
        rows[0] = l - mr; vals[0] = cscale(ph, s2i);
        rows[1] = l + mr; vals[1] = cscale(ph, ((mr & 1) ? -1.0 : 1.0) * s2i);
        return 2;
    }
    int ma = -mr;
    rows[0] = l - ma; vals[0] = cmul(ph, CD{0, -s2i});
    rows[1] = l + ma; vals[1] = cmul(ph, CD{0, ((ma & 1) ? -1.0 : 1.0) * s2i});
    return 2;
}

// Real-basis CG element; /sqrt(2lo+1) and *sqrt(2lo+1) in the reference cancel.
constexpr double real_cg_elem(int l1, int l2, int l3, int a, int b, int c) {
    int r1[2] = {0, 0}; CD v1[2] = {};
    int r2[2] = {0, 0}; CD v2[2] = {};
    int r3[2] = {0, 0}; CD v3[2] = {};
    int n1 = qcol(l1, a, r1, v1);
    int n2 = qcol(l2, b, r2, v2);
    int n3 = qcol(l3, c, r3, v3);
    CD acc{0, 0};
    for (int x = 0; x < n1; ++x)
        for (int y = 0; y < n2; ++y)
            for (int z = 0; z < n3; ++z) {
                double cg = su2_cg(l1, r1[x] - l1, l2, r2[y] - l2, l3, r3[z] - l3);
                if (cg != 0.0) {
                    CD t = cmul(cmul(v1[x], v2[y]), conjc(v3[z]));
                    acc.re += t.re * cg; acc.im += t.im * cg;
                }
            }
    return acc.re;
}

#define W_STRIDE 80   // 72 real output columns padded to 80 (5 N-tiles of 16)
#define W_ROWS   10   // K rows 0..8 real, row 9 kept all-zero (K padding target)

struct Tables {
    int nCG = 0;
    int pad_ = 0;
    int cgPack[1024] = {};       // (widx<<4) | j ; widx = kRow*80 + col
    float cgVal[1024] = {};
    unsigned colPack[80] = {};   // per GEMM column c: tOff | d3<<16 (c>=72 unused)
    unsigned grpPack[288] = {};  // per 4-element output group: dstbase | lo<<13
};

constexpr Tables buildTables() {
    Tables t{};
    // Enumerate CG paths exactly as the reference does.
    const int L[3] = {0, 1, 2};
    const int P1[3] = {1, -1, 1}, P2[3] = {1, -1, 1};
    int pl1[18] = {}, pl2[18] = {}, plo[18] = {}, ppo[18] = {};
    int nP = 0;
    for (int i = 0; i < 3; ++i)
        for (int j = 0; j < 3; ++j) {
            int l1 = L[i], l2 = L[j], po = P1[i] * P2[j];
            int lmin = l1 > l2 ? l1 - l2 : l2 - l1;
            for (int lo = lmin; lo <= l1 + l2; ++lo) {
                if (lo > 3) continue;
                if (lo == 0 && po != 1) continue;   // FILTER: (0,1),(1,±),(2,±),(3,±)
                pl1[nP] = l1; pl2[nP] = l2; plo[nP] = lo; ppo[nP] = po; ++nP;
            }
        }
    // Stable sort by e3nn key (lo, -po*(-1)^lo)
    int ord[18] = {};
    for (int k = 0; k < nP; ++k) ord[k] = k;
    for (int k = 1; k < nP; ++k) {
        int cur = ord[k];
        int k1 = plo[cur];
        int k2 = -ppo[cur] * ((plo[cur] & 1) ? -1 : 1);
        int m = k - 1;
        while (m >= 0) {
            int o = ord[m];
            int o1 = plo[o];
            int o2 = -ppo[o] * ((plo[o] & 1) ? -1 : 1);
            if (o1 > k1 || (o1 == k1 && o2 > k2)) { ord[m + 1] = ord[m]; --m; }
            else break;
        }
        ord[m + 1] = cur;
    }
    // Build sparse CG->W table + strip/output layout tables.
    // Strip is in OUTPUT-LINEAR order: t(u,c) = segT + u*d3 + k.
    int col = 0, outOff = 0, segT = 0, nCG = 0;
    for (int op = 0; op < nP; ++op) {
        int p = ord[op];
        int l1 = pl1[p], l2 = pl2[p], lo = plo[p];
        int d1 = 2 * l1 + 1, d2 = 2 * l2 + 1, d3 = 2 * lo + 1;
        int kb1 = (l1 == 0) ? 0 : (l1 == 1 ? 1 : 4);   // K-offset of l1 block in X1's 9 comps
        int jb2 = (l2 == 0) ? 0 : (l2 == 1 ? 1 : 4);   // offset of l2 block in x2's 9 comps
        for (int k = 0; k < d3; ++k) {
            int c = col + k;
            t.colPack[c] = (unsigned)(segT + k) | ((unsigned)d3 << 16);
            for (int a = 0; a < d1; ++a)
                for (int b = 0; b < d2; ++b) {
                    double v = real_cg_elem(l1, l2, lo, a, b, k);
                    if (v > 1e-9 || v < -1e-9) {
                        int widx = (kb1 + a) * W_STRIDE + c;
                        t.cgPack[nCG] = (widx << 4) | (jb2 + b);
                        t.cgVal[nCG] = (float)v;
                        ++nCG;
                    }
                }
        }
        // Segment length 16*d3 is a multiple of 16 -> 4-groups never straddle.
        for (int lt = 0; lt < 16 * d3; lt += 4)
            t.grpPack[(segT + lt) / 4] = (unsigned)(outOff + lt) | ((unsigned)lo << 13);
        col += d3;
        segT += 16 * d3;        // 1152 total
        outOff += 64 * d3;      // 4608 total
    }
    t.nCG = nCG;
    return t;
}

__device__ const Tables g_tab = buildTables();

// -------- gather one X1 element: (row u, gemm-K index k) -> 576-layout ------
// Branch-free K padding: load a clamped in-bounds address, select 0 after.
__device__ __forceinline__ float loadA(const float* __restrict__ x1b, int u, int k) {
    int kk = k <= 8 ? k : 8;
    int off = (kk == 0) ? u
            : (kk <= 3) ? (64 + u * 3 + (kk - 1))
                        : (256 + u * 5 + (kk - 4));
    float v = x1b[off];
    return (k <= 8) ? v : 0.0f;
}

// ------------------------------- kernel ------------------------------------

__global__ __launch_bounds__(256) void e3nn_tp_wmma_kernel(
    const float* __restrict__ x1, const float* __restrict__ x2,
    float* __restrict__ out, int batch)
{
    __shared__ float sW[8][W_ROWS * W_STRIDE];  // per-wave W_b (10x80)  25600 B
    __shared__ float sStrip[8][1152];           // per-wave out-linear   36864 B

    const int wave = threadIdx.x >> 5;
    const int lane = threadIdx.x & 31;
    const int b = blockIdx.x * 8 + wave;        // one sample per wave (wave-uniform)
    if (b >= batch) return;

    const float* x1b = x1 + (size_t)b * 576;
    const float* x2b = x2 + (size_t)b * 9;
    float* outb = out + (size_t)b * 4608;

    float* W = sW[wave];
    float* strip = sStrip[wave];

    // zero W (10x80): padded columns 72..79 AND padding row 9 stay zero
    for (int idx = lane; idx < W_ROWS * W_STRIDE; idx += 32) W[idx] = 0.0f;

    float xv[9];
    #pragma unroll
    for (int j = 0; j < 9; ++j) xv[j] = x2b[j];

    // W[i][c] += cg * x2[j]  (sparse; collisions resolved with ds_add_f32)
    const int nCG = g_tab.nCG;
    for (int e = lane; e < nCG; e += 32) {
        int pk = g_tab.cgPack[e];
        atomicAdd(&W[pk >> 4], g_tab.cgVal[e] * xv[pk & 15]);
    }

    const int half = lane >> 4;    // 0: lanes 0-15, 1: lanes 16-31
    const int ln = lane & 15;

    for (int mt = 0; mt < 4; ++mt) {
        const int u = mt * 16 + ln;
        // A fragments (16x4 f32 per K-step): VGPR0 holds K = 4ks+2h, VGPR1 K+1
        v2f afrag[3];
        #pragma unroll
        for (int ks = 0; ks < 3; ++ks) {
            int k0 = ks * 4 + half * 2;
            afrag[ks].x = loadA(x1b, u, k0);
            afrag[ks].y = loadA(x1b, u, k0 + 1);
        }
        #pragma unroll
        for (int nt = 0; nt < 5; ++nt) {
            const int n = nt * 16 + ln;
            v8f acc = {};
            #pragma unroll
            for (int ks = 0; ks < 3; ++ks) {
                int k0 = ks * 4 + half * 2;
                // clamp padded K rows onto the physical zero row 9 (branch-free)
                int r0 = k0 < 9 ? k0 : 9;
                int r1 = (k0 + 1) < 9 ? (k0 + 1) : 9;
                v2f bfrag;
                bfrag.x = W[r0 * W_STRIDE + n];
                bfrag.y = W[r1 * W_STRIDE + n];
                acc = __builtin_amdgcn_wmma_f32_16x16x4_f32(
                    false, afrag[ks], false, bfrag, (short)0, acc, false, false);
            }
            // D tile -> strip in output-linear order: t = tOff + M*d3
            if (n < 72) {
                unsigned cp = g_tab.colPack[n];
                int tOff = (int)(cp & 0xFFFFu);
                int d3 = (int)(cp >> 16);
                int base = tOff + (half * 8) * d3;
                #pragma unroll
                for (int r = 0; r < 8; ++r)
                    strip[base + r * d3] = acc[r];
            }
        }
        // Flush: strip is output-linear -> b128 LDS reads + b128 global stores
        for (int g = lane; g < 288; g += 32) {
            unsigned gp = g_tab.grpPack[g];
            int dstbase = (int)(gp & 0x1FFFu);
            int lo = (int)(gp >> 13);
            float4 v = *(const float4*)&strip[g * 4];
            *(float4*)(outb + dstbase + mt * 16 * (2 * lo + 1)) = v;
        }
    }
}

// ------------------------------- launcher ----------------------------------

extern "C" void kernel_launch(void* const* d_in, const int* in_sizes, int n_in,
                              void* d_out, int out_size, void* d_ws, size_t ws_size,
                              hipStream_t stream) {
    const float* x1 = (const float*)d_in[0];
    const float* x2 = (const float*)d_in[1];
    float* out = (float*)d_out;
    const int batch = in_sizes[0] / 576;
    const int grid = (batch + 7) / 8;
    e3nn_tp_wmma_kernel<<<grid, 256, 0, stream>>>(x1, x2, out, batch);
}